// TopKActivation_68324339745162
// MI455X (gfx1250) — compile-verified
//
#include <hip/hip_runtime.h>
#include <hip/hip_bf16.h>

// Top-K (k=64) per row of an 8192 x 16384 fp32 matrix, scattered back into a
// zero tensor. Memory-bound: stage each row in LDS via CDNA5 async
// global->LDS copies, radix-select the k-th largest key from LDS, then write
// the sparse row with non-temporal 128-bit stores.

#define TOPK_D       16384
#define TOPK_K       64
#define TOPK_THREADS 256

typedef __attribute__((ext_vector_type(4))) float v4f;
// Must match the builtin prototype's V4i: gcc-style vector, not ext_vector.
typedef int v4i_gcc __attribute__((vector_size(16)));

#define GLOBAL_AS __attribute__((address_space(1)))
#define LDS_AS    __attribute__((address_space(3)))

#if defined(__has_builtin)
#if __has_builtin(__builtin_amdgcn_global_load_async_to_lds_b128)
#define TOPK_HAVE_ASYNC_LDS 1
#endif
#endif

static __device__ __forceinline__ void topk_wait_async0() {
#if defined(__has_builtin)
#if __has_builtin(__builtin_amdgcn_s_wait_asynccnt)
  __builtin_amdgcn_s_wait_asynccnt(0);
#else
  asm volatile("s_wait_asynccnt 0" ::: "memory");
#endif
#else
  asm volatile("s_wait_asynccnt 0" ::: "memory");
#endif
}

// Monotonic float32 -> uint32 key: larger float => larger key.
static __device__ __forceinline__ unsigned int topk_key(float x) {
  unsigned int b = __float_as_uint(x);
  return (b & 0x80000000u) ? ~b : (b | 0x80000000u);
}

__global__ void
__launch_bounds__(TOPK_THREADS)
TopKActivation_68324339745162_kernel(const float* __restrict__ in,
                                     float* __restrict__ out) {
  extern __shared__ __align__(16) unsigned char smem[];
  float* row = (float*)smem;                    // TOPK_D floats = 64 KB (dynamic LDS)

  __shared__ unsigned int hist[256];
  __shared__ unsigned int s_prefix;
  __shared__ unsigned int s_rem;
  __shared__ unsigned int s_tie;

  const int tid = threadIdx.x;
  const long long r = blockIdx.x;
  const float* gin = in + r * (long long)TOPK_D;
  float* gout = out + r * (long long)TOPK_D;

  // ---- Stage the row into LDS (CDNA5 async copy; 16B per lane per op) ----
#if TOPK_HAVE_ASYNC_LDS
  {
    GLOBAL_AS char* gsrc = (GLOBAL_AS char*)gin;   // addrspacecast flat->global
    LDS_AS char* ldst = (LDS_AS char*)row;         // addrspacecast flat->lds
    for (int i = tid; i < TOPK_D / 4; i += TOPK_THREADS) {
      __builtin_amdgcn_global_load_async_to_lds_b128(
          (GLOBAL_AS v4i_gcc*)(gsrc + 16 * i),
          (LDS_AS v4i_gcc*)(ldst + 16 * i),
          /*imm offset*/ 0, /*cpol*/ 0);
    }
    topk_wait_async0();
  }
#else
  {
    const v4f* gv = (const v4f*)gin;
    v4f* rv = (v4f*)row;
    for (int i = tid; i < TOPK_D / 4; i += TOPK_THREADS) rv[i] = gv[i];
  }
#endif
  __syncthreads();

  // ---- Radix select: find key T of the 64th-largest element ----
  unsigned int prefix = 0;
  unsigned int rem = TOPK_K;

  for (int shift = 24; shift >= 0; shift -= 8) {
    hist[tid & 255] = 0;            // TOPK_THREADS == 256 bins
    __syncthreads();

    const unsigned int mask =
        (shift == 24) ? 0u : (0xFFFFFFFFu << (shift + 8));
    const unsigned int want = prefix & mask;

    const v4f* rv = (const v4f*)row;
    for (int i = tid; i < TOPK_D / 4; i += TOPK_THREADS) {
      v4f v = rv[i];
#pragma unroll
      for (int c = 0; c < 4; ++c) {
        unsigned int u = topk_key(v[c]);
        if ((u & mask) == want) {
          atomicAdd(&hist[(u >> shift) & 0xFFu], 1u);
        }
      }
    }
    __syncthreads();

    if (tid == 0) {
      unsigned int cum = 0;
      int b = 255;
      for (; b > 0; --b) {
        unsigned int c = hist[b];
        if (cum + c >= rem) break;
        cum += c;
      }
      s_prefix = prefix | ((unsigned int)b << shift);
      s_rem = rem - cum;
    }
    __syncthreads();
    prefix = s_prefix;
    rem = s_rem;
    __syncthreads();
  }

  const unsigned int T = prefix;   // key of the k-th largest
  if (tid == 0) s_tie = 0;
  __syncthreads();

  // ---- Scatter: keep values with key > T, plus `rem` ties at key == T ----
  {
    const v4f* rv = (const v4f*)row;
    v4f* gv = (v4f*)gout;
    for (int i = tid; i < TOPK_D / 4; i += TOPK_THREADS) {
      v4f v = rv[i];
      v4f o;
#pragma unroll
      for (int c = 0; c < 4; ++c) {
        float x = v[c];
        unsigned int u = topk_key(x);
        float keep = 0.0f;
        if (u > T) {
          keep = x;
        } else if (u == T) {
          unsigned int slot = atomicAdd(&s_tie, 1u);
          if (slot < rem) keep = x;
        }
        o[c] = keep;
      }
      __builtin_nontemporal_store(o, gv + i);
    }
  }
}

extern "C" void kernel_launch(void* const* d_in, const int* in_sizes, int n_in,
                              void* d_out, int out_size, void* d_ws, size_t ws_size,
                              hipStream_t stream) {
  const float* in = (const float*)d_in[0];
  // d_in[1] is k (==64 per setup_inputs); compile-time constant here.
  float* out = (float*)d_out;

  const int rows = in_sizes[0] / TOPK_D;   // 8192
  dim3 grid((unsigned)rows), block(TOPK_THREADS);
  const size_t shmem = (size_t)TOPK_D * sizeof(float);  // 64 KB dynamic LDS

  hipLaunchKernelGGL(TopKActivation_68324339745162_kernel, grid, block, shmem,
                     stream, in, out);
}